// SplineCNN_82231443849688
// MI455X (gfx1250) — compile-verified
//
#include <hip/hip_runtime.h>
#include <hip/hip_bf16.h>
#include <math.h>

typedef __attribute__((ext_vector_type(2))) float v2f;
typedef __attribute__((ext_vector_type(8))) float v8f;

// ---------------------------------------------------------------------------
// Zero a float region (ws is poisoned 0xAA; must re-zero every call).
// ---------------------------------------------------------------------------
__global__ void zero_kernel(float* __restrict__ p, long long n) {
    long long i = (long long)blockIdx.x * blockDim.x + threadIdx.x;
    if (i < n) p[i] = 0.0f;
}

// ---------------------------------------------------------------------------
// GEMM1: y1r1[n, 0:16]  = x[n,:] @ W1[1]   (128x16)
//        y1r1[n,16:32]  = x[n,:] @ root1   (128x16)
// One wave computes a 16-row x 32-col tile using V_WMMA_F32_16X16X4_F32.
// A frag (16x4 f32): lanes 0-15 hold K=0,1 in v0,v1; lanes 16-31 hold K=2,3.
// B frag (4x16 f32): lane = column, same K split across lane halves.
// D (16x16 f32, 8 VGPRs): VGPR r -> rows r (lanes 0-15) and r+8 (lanes 16-31).
// ---------------------------------------------------------------------------
__global__ void gemm1_kernel(const float* __restrict__ x,
                             const float* __restrict__ w1k1,   // 128x16
                             const float* __restrict__ root1,  // 128x16
                             float* __restrict__ y1r1, int N) {
    const int lane = threadIdx.x & 31;
    const int wave = threadIdx.x >> 5;
    const int tile = blockIdx.x * 4 + wave;
    const int row0 = tile * 16;
    if (row0 >= N) return;

    const int colL = lane & 15;
    const int kSub = (lane >> 4) * 2;           // 0 for lanes 0-15, 2 for 16-31
    int rowA = row0 + colL;                     // A-matrix row for this lane
    if (rowA >= N) rowA = N - 1;                // safe duplicate (stores guarded)

    v8f c0 = {};  // x @ W1[1]
    v8f c1 = {};  // x @ root1
#pragma unroll 4
    for (int kk = 0; kk < 32; ++kk) {
        const int k = kk * 4 + kSub;
        v2f a;
        a.x = x[(long long)rowA * 128 + k];
        a.y = x[(long long)rowA * 128 + k + 1];
        v2f b0;
        b0.x = w1k1[k * 16 + colL];
        b0.y = w1k1[(k + 1) * 16 + colL];
        v2f b1;
        b1.x = root1[k * 16 + colL];
        b1.y = root1[(k + 1) * 16 + colL];
        c0 = __builtin_amdgcn_wmma_f32_16x16x4_f32(false, a, false, b0,
                                                   (short)0, c0, false, false);
        c1 = __builtin_amdgcn_wmma_f32_16x16x4_f32(false, a, false, b1,
                                                   (short)0, c1, false, false);
    }

    const int rbase = (lane >> 4) * 8;
#pragma unroll
    for (int r = 0; r < 8; ++r) {
        const int row = row0 + rbase + r;
        if (row < N) {
            y1r1[(long long)row * 32 + colL]      = c0[r];
            y1r1[(long long)row * 32 + 16 + colL] = c1[r];
        }
    }
}

// ---------------------------------------------------------------------------
// Edge scatter 1: agg1[dst,c] += y1[src,c]; deg[dst] += 1.
// 16 lanes per edge (lane = channel) -> coalesced gather + coalesced atomics.
// ---------------------------------------------------------------------------
__global__ void scatter1_kernel(const int* __restrict__ ei, int E,
                                const float* __restrict__ y1r1,
                                float* __restrict__ agg1,
                                float* __restrict__ deg) {
    long long gid = (long long)blockIdx.x * blockDim.x + threadIdx.x;
    int e = (int)(gid >> 4);
    int c = (int)(gid & 15);
    if (e >= E) return;
    int src = ei[e];
    int dst = ei[E + e];
    atomicAdd(&agg1[(long long)dst * 16 + c], y1r1[(long long)src * 32 + c]);
    if (c == 0) atomicAdd(&deg[dst], 1.0f);
}

// ---------------------------------------------------------------------------
// h = elu(agg1/max(deg,1) + r1 + b1)
// ---------------------------------------------------------------------------
__global__ void combine1_kernel(const float* __restrict__ agg1,
                                const float* __restrict__ deg,
                                const float* __restrict__ y1r1,
                                const float* __restrict__ b1,
                                float* __restrict__ h, int N) {
    long long gid = (long long)blockIdx.x * blockDim.x + threadIdx.x;
    if (gid >= (long long)N * 16) return;
    int n = (int)(gid >> 4);
    int c = (int)(gid & 15);
    float v = agg1[gid] / fmaxf(deg[n], 1.0f)
            + y1r1[(long long)n * 32 + 16 + c] + b1[c];
    h[gid] = (v > 0.0f) ? v : (expf(v) - 1.0f);
}

// ---------------------------------------------------------------------------
// Edge scatter 2: agg2[dst,c] += h[src,c]   (aggregate BEFORE W2 — linear).
// ---------------------------------------------------------------------------
__global__ void scatter2_kernel(const int* __restrict__ ei, int E,
                                const float* __restrict__ h,
                                float* __restrict__ agg2) {
    long long gid = (long long)blockIdx.x * blockDim.x + threadIdx.x;
    int e = (int)(gid >> 4);
    int c = (int)(gid & 15);
    if (e >= E) return;
    int src = ei[e];
    int dst = ei[E + e];
    atomicAdd(&agg2[(long long)dst * 16 + c], h[(long long)src * 16 + c]);
}

// ---------------------------------------------------------------------------
// GEMM2: out[n,:] = (agg2[n]/max(deg,1)) @ W2[1] + h[n] @ root2 + b2
// Fused as [agg2*invd | h] (N x 32) @ [[W2[1]];[root2]] (32 x 40).
// 3 column tiles of 16 (cols 40..47 masked). Writes pre-softmax into d_out.
// ---------------------------------------------------------------------------
__device__ __forceinline__ float b2load(const float* w2k1, const float* root2,
                                        int k, int col) {
    if (col >= 40) return 0.0f;
    return (k < 16) ? w2k1[k * 40 + col] : root2[(k - 16) * 40 + col];
}

__global__ void gemm2_kernel(const float* __restrict__ agg2,
                             const float* __restrict__ deg,
                             const float* __restrict__ h,
                             const float* __restrict__ w2k1,   // 16x40
                             const float* __restrict__ root2,  // 16x40
                             const float* __restrict__ b2,     // 40
                             float* __restrict__ out, int N) {
    const int lane = threadIdx.x & 31;
    const int wave = threadIdx.x >> 5;
    const int tile = blockIdx.x * 4 + wave;
    const int row0 = tile * 16;
    if (row0 >= N) return;

    const int colL = lane & 15;
    const int kSub = (lane >> 4) * 2;
    int rowA = row0 + colL;
    if (rowA >= N) rowA = N - 1;
    const float invd = 1.0f / fmaxf(deg[rowA], 1.0f);

    v8f c0 = {}, c1 = {}, c2 = {};
#pragma unroll
    for (int kk = 0; kk < 8; ++kk) {
        const int k = kk * 4 + kSub;  // k is even: k and k+1 stay in same half
        v2f a;
        a.x = (k < 16) ? agg2[(long long)rowA * 16 + k] * invd
                       : h[(long long)rowA * 16 + (k - 16)];
        a.y = (k + 1 < 16) ? agg2[(long long)rowA * 16 + k + 1] * invd
                           : h[(long long)rowA * 16 + (k + 1 - 16)];
        v2f b0, b1, b2v;
        b0.x  = b2load(w2k1, root2, k,     colL);
        b0.y  = b2load(w2k1, root2, k + 1, colL);
        b1.x  = b2load(w2k1, root2, k,     16 + colL);
        b1.y  = b2load(w2k1, root2, k + 1, 16 + colL);
        b2v.x = b2load(w2k1, root2, k,     32 + colL);
        b2v.y = b2load(w2k1, root2, k + 1, 32 + colL);
        c0 = __builtin_amdgcn_wmma_f32_16x16x4_f32(false, a, false, b0,
                                                   (short)0, c0, false, false);
        c1 = __builtin_amdgcn_wmma_f32_16x16x4_f32(false, a, false, b1,
                                                   (short)0, c1, false, false);
        c2 = __builtin_amdgcn_wmma_f32_16x16x4_f32(false, a, false, b2v,
                                                   (short)0, c2, false, false);
    }

    const int rbase = (lane >> 4) * 8;
#pragma unroll
    for (int r = 0; r < 8; ++r) {
        const int row = row0 + rbase + r;
        if (row >= N) continue;
        out[(long long)row * 40 + colL]      = c0[r] + b2[colL];
        out[(long long)row * 40 + 16 + colL] = c1[r] + b2[16 + colL];
        if (colL < 8)
            out[(long long)row * 40 + 32 + colL] = c2[r] + b2[32 + colL];
    }
}

// ---------------------------------------------------------------------------
// log_softmax over 40 classes, in-place on d_out. One thread per node.
// ---------------------------------------------------------------------------
__global__ void logsoftmax_kernel(float* __restrict__ out, int N) {
    int n = blockIdx.x * blockDim.x + threadIdx.x;
    if (n >= N) return;
    float v[40];
    float m = -INFINITY;
#pragma unroll
    for (int i = 0; i < 40; ++i) {
        v[i] = out[(long long)n * 40 + i];
        m = fmaxf(m, v[i]);
    }
    float s = 0.0f;
#pragma unroll
    for (int i = 0; i < 40; ++i) s += expf(v[i] - m);
    float ls = logf(s);
#pragma unroll
    for (int i = 0; i < 40; ++i) out[(long long)n * 40 + i] = v[i] - m - ls;
}

// ---------------------------------------------------------------------------
// Host launcher. Inputs: x(N*128), edge_index(2*E), W1(2*128*16), root1(128*16),
// b1(16), W2(2*16*40), root2(16*40), b2(40). Output: N*40 f32 log-probs.
// ---------------------------------------------------------------------------
extern "C" void kernel_launch(void* const* d_in, const int* in_sizes, int n_in,
                              void* d_out, int out_size, void* d_ws, size_t ws_size,
                              hipStream_t stream) {
    const float* x     = (const float*)d_in[0];
    const int*   ei    = (const int*)d_in[1];
    const float* W1    = (const float*)d_in[2];
    const float* root1 = (const float*)d_in[3];
    const float* b1    = (const float*)d_in[4];
    const float* W2    = (const float*)d_in[5];
    const float* root2 = (const float*)d_in[6];
    const float* b2    = (const float*)d_in[7];
    float*       out   = (float*)d_out;

    const int N = in_sizes[0] / 128;
    const int E = in_sizes[1] / 2;

    // B-spline with ones pseudo-coords => only kernel matrix [1] contributes.
    const float* w1k1 = W1 + 128 * 16;  // W1[1]
    const float* w2k1 = W2 + 16 * 40;   // W2[1]

    // Workspace layout (floats): agg1[N*16] | deg[N] | agg2[N*16] | y1r1[N*32] | h[N*16]
    float* ws   = (float*)d_ws;
    float* agg1 = ws;
    float* deg  = ws + (size_t)N * 16;
    float* agg2 = ws + (size_t)N * 17;
    float* y1r1 = ws + (size_t)N * 33;
    float* h    = ws + (size_t)N * 65;

    // 1) zero accumulators (agg1|deg|agg2 are contiguous: N*33 floats)
    long long nz = (long long)N * 33;
    zero_kernel<<<(unsigned)((nz + 255) / 256), 256, 0, stream>>>(ws, nz);

    const int tiles   = (N + 15) / 16;
    const int gblocks = (tiles + 3) / 4;  // 4 waves (128 thr) per block

    // 2) y1r1 = x @ [W1[1] | root1]
    gemm1_kernel<<<gblocks, 128, 0, stream>>>(x, w1k1, root1, y1r1, N);

    // 3) agg1[dst] += y1[src]; deg[dst] += 1
    long long sthreads = (long long)E * 16;
    unsigned sblocks = (unsigned)((sthreads + 255) / 256);
    scatter1_kernel<<<sblocks, 256, 0, stream>>>(ei, E, y1r1, agg1, deg);

    // 4) h = elu(agg1/deg + r1 + b1)
    long long cthreads = (long long)N * 16;
    combine1_kernel<<<(unsigned)((cthreads + 255) / 256), 256, 0, stream>>>(
        agg1, deg, y1r1, b1, h, N);

    // 5) agg2[dst] += h[src]
    scatter2_kernel<<<sblocks, 256, 0, stream>>>(ei, E, h, agg2);

    // 6) out = [agg2/deg | h] @ [W2[1]; root2] + b2
    gemm2_kernel<<<gblocks, 128, 0, stream>>>(agg2, deg, h, w2k1, root2, b2, out, N);

    // 7) log_softmax in place
    logsoftmax_kernel<<<(unsigned)((N + 255) / 256), 256, 0, stream>>>(out, N);
}